// MultiTaskLossWrapper_29420525978037
// MI455X (gfx1250) — compile-verified
//
#include <hip/hip_runtime.h>
#include <math.h>

// ---------------------------------------------------------------------------
// MI455X / gfx1250 implementation.
//   cross GEMM  : V_WMMA_F32_16X16X4_F32, 3 accumulating steps (D padded to 12)
//   D_norm tile : resident in LDS (16 cols x 2048 rows f32 ~ 128KB; CDNA5 WGP
//                 has 320KB LDS so a full column fits on-chip)
//   top-64      : wave32 extract-max, 64 values/lane in VGPRs, deterministic
// ---------------------------------------------------------------------------

typedef __attribute__((ext_vector_type(2))) float v2f;
typedef __attribute__((ext_vector_type(8))) float v8f;

#define BB 8
#define NN 2048
#define MM 2048
#define DD 9
#define DP 12          // D padded to 12 = 3 x K4 WMMA steps
#define KK 64          // top-k
#define NEG_INF (-3.402823466e38f)
#define LSTRIDE 2052   // 2048 + 4 pad -> lane bank spread (stride%64 == 4)

// workspace layout (float offsets)
#define WS_MU     16                      // BB*16
#define WS_C      160                     // BB*81
#define WS_CSYM   960                     // BB*81
#define WS_QA     2048                    // BB*MM
#define WS_QB     (WS_QA + BB*MM)         // BB*NN
#define WS_AP     (WS_QB + BB*NN)         // BB*MM*DP  (a @ Csym, zero-padded)
#define WS_BP     (WS_AP + BB*MM*DP)      // BB*NN*DP  (b, zero-padded)
#define WS_COLSUM (WS_BP + BB*NN*DP)      // BB*NN     (per-column top-k sums)

// ---------------------------------------------------------------------------
// K1: per-batch mu, cov = yc^T yc, C = pinv(cov), Csym = C + C^T
// ---------------------------------------------------------------------------
__global__ void __launch_bounds__(256) k_stats(const float* __restrict__ targets,
                                               float* __restrict__ ws)
{
    const int b    = blockIdx.x;
    const int tid  = threadIdx.x;
    const int lane = tid & 31;
    const int w    = tid >> 5;
    const float* T = targets + (size_t)b * MM * DD;

    __shared__ float wavebuf[8][81];
    __shared__ float muS[9];
    __shared__ float covS[81];

    // ---- mean over M (deterministic shuffle-tree + fixed-order wave merge)
    float ms[9];
#pragma unroll
    for (int d = 0; d < 9; ++d) ms[d] = 0.f;
    for (int m = tid; m < MM; m += 256) {
#pragma unroll
        for (int d = 0; d < 9; ++d) ms[d] += T[m * DD + d];
    }
#pragma unroll
    for (int d = 0; d < 9; ++d) {
        float s = ms[d];
#pragma unroll
        for (int off = 16; off > 0; off >>= 1) s += __shfl_xor(s, off, 32);
        if (lane == 0) wavebuf[w][d] = s;
    }
    __syncthreads();
    if (tid < 9) {
        float s = 0.f;
#pragma unroll
        for (int i = 0; i < 8; ++i) s += wavebuf[i][tid];
        muS[tid] = s / (float)MM;
    }
    __syncthreads();

    // ---- cov = yc^T yc  (81 per-thread register accumulators, fully unrolled)
    float cv[81];
#pragma unroll
    for (int e = 0; e < 81; ++e) cv[e] = 0.f;
    for (int m = tid; m < MM; m += 256) {
        float yc[9];
#pragma unroll
        for (int d = 0; d < 9; ++d) yc[d] = T[m * DD + d] - muS[d];
#pragma unroll
        for (int d = 0; d < 9; ++d)
#pragma unroll
            for (int e = 0; e < 9; ++e) cv[d * 9 + e] += yc[d] * yc[e];
    }
#pragma unroll
    for (int e = 0; e < 81; ++e) {
        float s = cv[e];
#pragma unroll
        for (int off = 16; off > 0; off >>= 1) s += __shfl_xor(s, off, 32);
        if (lane == 0) wavebuf[w][e] = s;
    }
    __syncthreads();
    if (tid < 81) {
        float s = 0.f;
#pragma unroll
        for (int i = 0; i < 8; ++i) s += wavebuf[i][tid];
        covS[tid] = s;
    }
    __syncthreads();

    // ---- one lane: pinv(cov) = inv(cov^T cov) cov^T via Gauss-Jordan
    if (tid == 0) {
        float G[81], Inv[81], Cm[81];
        for (int i = 0; i < 9; ++i)
            for (int j = 0; j < 9; ++j) {
                float s = 0.f;
                for (int l = 0; l < 9; ++l) s += covS[l * 9 + i] * covS[l * 9 + j];
                G[i * 9 + j] = s;                       // (cov^T cov)
            }
        for (int i = 0; i < 81; ++i) Inv[i] = 0.f;
        for (int i = 0; i < 9; ++i) Inv[i * 9 + i] = 1.f;
        for (int col = 0; col < 9; ++col) {             // partial pivoting
            int piv = col; float pv = fabsf(G[col * 9 + col]);
            for (int r = col + 1; r < 9; ++r) {
                float a = fabsf(G[r * 9 + col]);
                if (a > pv) { pv = a; piv = r; }
            }
            if (piv != col) {
                for (int j = 0; j < 9; ++j) {
                    float t = G[col * 9 + j]; G[col * 9 + j] = G[piv * 9 + j]; G[piv * 9 + j] = t;
                    t = Inv[col * 9 + j]; Inv[col * 9 + j] = Inv[piv * 9 + j]; Inv[piv * 9 + j] = t;
                }
            }
            float d = G[col * 9 + col];
            float rinv = (d != 0.f) ? 1.f / d : 0.f;
            for (int j = 0; j < 9; ++j) { G[col * 9 + j] *= rinv; Inv[col * 9 + j] *= rinv; }
            for (int r = 0; r < 9; ++r) if (r != col) {
                float f = G[r * 9 + col];
                for (int j = 0; j < 9; ++j) {
                    G[r * 9 + j]  -= f * G[col * 9 + j];
                    Inv[r * 9 + j] -= f * Inv[col * 9 + j];
                }
            }
        }
        for (int i = 0; i < 9; ++i)                     // C = Inv * cov^T
            for (int j = 0; j < 9; ++j) {
                float s = 0.f;
                for (int l = 0; l < 9; ++l) s += Inv[i * 9 + l] * covS[j * 9 + l];
                Cm[i * 9 + j] = s;
            }
        float* mu = ws + WS_MU + b * 16;
        float* Cw = ws + WS_C + b * 81;
        float* Cs = ws + WS_CSYM + b * 81;
        for (int d = 0; d < 9; ++d) mu[d] = muS[d];
        for (int i = 0; i < 9; ++i)
            for (int j = 0; j < 9; ++j) {
                Cw[i * 9 + j] = Cm[i * 9 + j];
                Cs[i * 9 + j] = Cm[i * 9 + j] + Cm[j * 9 + i];
            }
    }
}

// ---------------------------------------------------------------------------
// K2: per row r (M == N): qa, qb, aPrime = yc@Csym (padded), bPad (padded)
// ---------------------------------------------------------------------------
__global__ void __launch_bounds__(256) k_prep(const float* __restrict__ outputs,
                                              const float* __restrict__ targets,
                                              float* __restrict__ ws)
{
    int i = blockIdx.x * blockDim.x + threadIdx.x;
    if (i >= BB * MM) return;
    int b = i / MM, r = i % MM;
    const float* mu = ws + WS_MU + b * 16;
    const float* C  = ws + WS_C + b * 81;
    const float* Cs = ws + WS_CSYM + b * 81;

    float yc[9];
#pragma unroll
    for (int d = 0; d < 9; ++d) yc[d] = targets[(size_t)(b * MM + r) * DD + d] - mu[d];
    float qa = 0.f;
#pragma unroll
    for (int d = 0; d < 9; ++d) {
        float cd = 0.f;
#pragma unroll
        for (int e = 0; e < 9; ++e) cd += C[d * 9 + e] * yc[e];
        qa += yc[d] * cd;
    }
    ws[WS_QA + b * MM + r] = qa;
    float* ap = ws + WS_AP + (size_t)(b * MM + r) * DP;
#pragma unroll
    for (int d = 0; d < 9; ++d) {
        float s = 0.f;
#pragma unroll
        for (int e = 0; e < 9; ++e) s += yc[e] * Cs[e * 9 + d];
        ap[d] = s;
    }
    ap[9] = 0.f; ap[10] = 0.f; ap[11] = 0.f;

    float bv[9];
#pragma unroll
    for (int d = 0; d < 9; ++d) bv[d] = outputs[(size_t)(b * NN + r) * DD + d];
    float qb = 0.f;
#pragma unroll
    for (int d = 0; d < 9; ++d) {
        float cd = 0.f;
#pragma unroll
        for (int e = 0; e < 9; ++e) cd += C[d * 9 + e] * bv[e];
        qb += bv[d] * cd;
    }
    ws[WS_QB + b * NN + r] = qb;
    float* bp = ws + WS_BP + (size_t)(b * NN + r) * DP;
#pragma unroll
    for (int d = 0; d < 9; ++d) bp[d] = bv[d];
    bp[9] = 0.f; bp[10] = 0.f; bp[11] = 0.f;
}

// ---------------------------------------------------------------------------
// K3: one workgroup = one (batch, 16-wide n-tile).
//     Phase 1: 8 waves build D_norm columns in LDS via V_WMMA_F32_16X16X4_F32
//     Phase 2: each wave does top-64 extract-max on 2 columns (regs only)
// ---------------------------------------------------------------------------
__global__ void __launch_bounds__(256) k_cross_topk(float* __restrict__ ws)
{
    extern __shared__ float smem[];                 // 16 * LSTRIDE floats
    const int blk  = blockIdx.x;
    const int b    = blk >> 7;                      // 128 n-tiles per batch
    const int nt   = (blk & 127) << 4;
    const int tid  = threadIdx.x;
    const int lane = tid & 31, w = tid >> 5;
    const int hi   = lane >> 4, lo = lane & 15;

    const float* AP = ws + WS_AP + (size_t)b * MM * DP;
    const float* BP = ws + WS_BP + (size_t)b * NN * DP;
    const float* QA = ws + WS_QA + b * MM;
    const float* QB = ws + WS_QB + b * NN;

    const int   n   = nt + lo;
    const float qbn = QB[n];

    // B fragments (4x16 f32 layout: lanes 0-15 hold K={k0,k0+1}, 16-31 K={k0+2,k0+3})
    v2f bf[3];
#pragma unroll
    for (int j = 0; j < 3; ++j) {
        float2 t = *reinterpret_cast<const float2*>(BP + (size_t)n * DP + 4 * j + 2 * hi);
        bf[j].x = t.x; bf[j].y = t.y;
    }

    // Phase 1: 2048/16 = 128 m-tiles, round-robin over 8 waves (uniform trip count)
    for (int mt = w * 16; mt < MM; mt += 8 * 16) {
        const int arow = mt + lo;                   // A-matrix row M = lane&15
        v8f c = {0.f, 0.f, 0.f, 0.f, 0.f, 0.f, 0.f, 0.f};
#pragma unroll
        for (int j = 0; j < 3; ++j) {               // K = 0..11 in 3 x K4 steps
            float2 t = *reinterpret_cast<const float2*>(AP + (size_t)arow * DP + 4 * j + 2 * hi);
            v2f a; a.x = t.x; a.y = t.y;
            c = __builtin_amdgcn_wmma_f32_16x16x4_f32(false, a, false, bf[j],
                                                      (short)0, c, false, false);
        }
        // C/D layout: lane holds 8 rows m = mt + r + 8*hi of column n = nt+lo
#pragma unroll
        for (int r = 0; r < 8; ++r) {
            int m = mt + r + 8 * hi;
            smem[lo * LSTRIDE + m] = QA[m] + qbn - c[r];
        }
    }
    __syncthreads();

    // Phase 2: wave w owns columns 2w and 2w+1; 64 values/lane kept in VGPRs
    float* colsum = ws + WS_COLSUM;
#pragma unroll 1
    for (int cc = 0; cc < 2; ++cc) {
        const int col = w * 2 + cc;
        float v[64];
#pragma unroll
        for (int i = 0; i < 64; ++i) v[i] = smem[col * LSTRIDE + lane + 32 * i];
        float sum = 0.f;
#pragma unroll 1
        for (int it = 0; it < KK; ++it) {           // 64 deterministic extract-max
            float lm = NEG_INF;
#pragma unroll
            for (int i = 0; i < 64; ++i) lm = fmaxf(lm, v[i]);
            float wm = lm;
#pragma unroll
            for (int off = 16; off > 0; off >>= 1) wm = fmaxf(wm, __shfl_xor(wm, off, 32));
            sum += wm;
            unsigned long long ball = __ballot(lm == wm);
            int leader = __ffsll(ball) - 1;
            if (lane == leader) {                   // remove exactly one instance
                bool rem = false;
#pragma unroll
                for (int i = 0; i < 64; ++i) {
                    if (!rem && v[i] == wm) { v[i] = NEG_INF; rem = true; }
                }
            }
        }
        if (lane == 0) colsum[b * NN + nt + col] = sum;
    }
}

// ---------------------------------------------------------------------------
// K4: deterministic fixed-tree reduction of the 16384 column sums -> mean
// ---------------------------------------------------------------------------
__global__ void __launch_bounds__(256) k_final(const float* __restrict__ ws,
                                               float* __restrict__ out)
{
    __shared__ float buf[256];
    int tid = threadIdx.x;
    const float* cs = ws + WS_COLSUM;
    float s = 0.f;
    for (int i = tid; i < BB * NN; i += 256) s += cs[i];
    buf[tid] = s;
    __syncthreads();
    for (int st = 128; st > 0; st >>= 1) {
        if (tid < st) buf[tid] += buf[tid + st];
        __syncthreads();
    }
    if (tid == 0) out[0] = buf[0] / (float)(BB * NN * KK);
}

// ---------------------------------------------------------------------------
extern "C" void kernel_launch(void* const* d_in, const int* in_sizes, int n_in,
                              void* d_out, int out_size, void* d_ws, size_t ws_size,
                              hipStream_t stream)
{
    const float* outputs = (const float*)d_in[0];   // (B,N,9)
    const float* targets = (const float*)d_in[1];   // (B,M,9)
    float* ws  = (float*)d_ws;
    float* out = (float*)d_out;

    k_stats<<<BB, 256, 0, stream>>>(targets, ws);
    k_prep<<<(BB * MM + 255) / 256, 256, 0, stream>>>(outputs, targets, ws);
    size_t smem = (size_t)16 * LSTRIDE * sizeof(float);   // ~128KB of 320KB/WGP LDS
    k_cross_topk<<<BB * (NN / 16), 256, smem, stream>>>(ws);
    k_final<<<1, 256, 0, stream>>>(ws, out);
}